// GraphSAGE_42812234006861
// MI455X (gfx1250) — compile-verified
//
#include <hip/hip_runtime.h>
#include <hip/hip_bf16.h>

typedef float v2f __attribute__((ext_vector_type(2)));
typedef float v8f __attribute__((ext_vector_type(8)));

#define GN    50000
#define GE    600000
#define DIN_  128
#define H_    128
#define DOUT_ 64
#define LPAD  132   // LDS row stride in floats (bank-conflict-free, 8/16B aligned)

// ---------------------------------------------------------------------------
// Zero a float buffer (float4 vectorized; count must be divisible by 4).
// ---------------------------------------------------------------------------
__global__ void zero_f32(float4* __restrict__ p, int count4) {
    int i = blockIdx.x * blockDim.x + threadIdx.x;
    if (i < count4) p[i] = make_float4(0.f, 0.f, 0.f, 0.f);
}

// ---------------------------------------------------------------------------
// Degree: deg[dst] += 1 per edge (hardware f32 atomic -> global_atomic_add_f32)
// ---------------------------------------------------------------------------
__global__ void degree_kernel(const long long* __restrict__ eidx,
                              float* __restrict__ deg, int E) {
    int e = blockIdx.x * blockDim.x + threadIdx.x;
    if (e < E) {
        int dst = (int)eidx[(size_t)E + e];
        unsafeAtomicAdd(&deg[dst], 1.0f);
    }
}

// ---------------------------------------------------------------------------
// Scatter-accumulate: agg[dst, :] += feat[src, :].  One wave32 per edge.
// Each lane: one coalesced float4 load (512B per wave) + 4 f32 atomics.
// ---------------------------------------------------------------------------
__global__ void scatter_accum(const float* __restrict__ feat,
                              const long long* __restrict__ eidx,
                              float* __restrict__ agg, int E) {
    int gtid = blockIdx.x * blockDim.x + threadIdx.x;
    int e    = gtid >> 5;
    int lane = gtid & 31;
    if (e >= E) return;
    int src = (int)eidx[e];
    int dst = (int)eidx[(size_t)E + e];
    const float4 v = *(const float4*)(feat + (size_t)src * DIN_ + lane * 4);
    float* ap = agg + (size_t)dst * DIN_ + lane * 4;
    unsafeAtomicAdd(ap + 0, v.x);
    unsafeAtomicAdd(ap + 1, v.y);
    unsafeAtomicAdd(ap + 2, v.z);
    unsafeAtomicAdd(ap + 3, v.w);
}

// ---------------------------------------------------------------------------
// Fused SAGE layer:  out = act( (agg/deg) @ wl + x @ wr + b )
//
// Block = NOUT/16 waves; block owns a 16-row M tile. The two A operands
// (mean = agg/deg, and x) are staged once per block into LDS with a fully
// coalesced float4 sweep (1/deg folded in at stage time), then every wave
// computes one 16x16 N-tile: K-loop of 32 steps, two
// v_wmma_f32_16x16x4_f32 per step sharing the f32 accumulator.
//
// VGPR layouts (ISA 7.12.2, 16x16x4 f32):
//   A (16x4):  lanes 0-15 -> M=0..15; vgpr0 = K {0 | 2}, vgpr1 = K {1 | 3}
//   B (4x16):  lanes 0-15 -> N=0..15; vgpr0 = K {0 | 2}, vgpr1 = K {1 | 3}
//   C (16x16): vgpr r: lanes 0-15 -> (M=r, N=lane), lanes 16-31 -> (M=r+8)
// ---------------------------------------------------------------------------
template <int NOUT, bool RELU>
__global__ __launch_bounds__(32 * (NOUT / 16))
void sage_gemm(const float* __restrict__ agg, const float* __restrict__ deg,
               const float* __restrict__ xin, const float* __restrict__ wl,
               const float* __restrict__ wr,  const float* __restrict__ bias,
               float* __restrict__ out) {
    __shared__ float smean[16 * LPAD];
    __shared__ float sx[16 * LPAD];

    const int tid  = threadIdx.x;
    const int lane = tid & 31;
    const int wave = tid >> 5;
    const int row0 = blockIdx.x * 16;      // M tile (50000 = 3125*16 exact)
    const int n0   = wave * 16;            // N tile within output width
    const int col  = lane & 15;
    const int khalf = (lane >> 4) << 1;    // 0 for lanes 0-15, 2 for 16-31

    // ---- Stage A tiles (16 x 128, two matrices) into LDS, prescaled ----
    constexpr int NT = 32 * (NOUT / 16);
#pragma unroll
    for (int j = tid; j < 16 * 32; j += NT) {     // 512 float4s per matrix
        const int row  = j >> 5;
        const int c4   = (j & 31) << 2;
        const int grow = row0 + row;
        const float dsc = 1.0f / fmaxf(deg[grow], 1.0f);
        const float4 av = *(const float4*)(agg + (size_t)grow * H_ + c4);
        const float4 xv = *(const float4*)(xin + (size_t)grow * H_ + c4);
        *(float4*)(smean + row * LPAD + c4) =
            make_float4(av.x * dsc, av.y * dsc, av.z * dsc, av.w * dsc);
        *(float4*)(sx + row * LPAD + c4) = xv;
    }
    __syncthreads();

    // Accumulator starts at the (per-column) bias.
    const float bv = bias[n0 + col];
    v8f c;
#pragma unroll
    for (int i = 0; i < 8; ++i) c[i] = bv;

    const float* amp = smean + col * LPAD;  // this lane's A row (M = col)
    const float* axp = sx    + col * LPAD;

#pragma unroll 4
    for (int k = 0; k < H_; k += 4) {
        const int ka = k + khalf;
        const v2f a_mean = *(const v2f*)(amp + ka);   // ds_load_b64
        const v2f a_root = *(const v2f*)(axp + ka);
        v2f b_l, b_r;
        b_l.x = wl[(size_t)ka * NOUT + n0 + col];
        b_l.y = wl[(size_t)(ka + 1) * NOUT + n0 + col];
        b_r.x = wr[(size_t)ka * NOUT + n0 + col];
        b_r.y = wr[(size_t)(ka + 1) * NOUT + n0 + col];
        c = __builtin_amdgcn_wmma_f32_16x16x4_f32(false, a_mean, false, b_l,
                                                  (short)0, c, false, false);
        c = __builtin_amdgcn_wmma_f32_16x16x4_f32(false, a_root, false, b_r,
                                                  (short)0, c, false, false);
    }

    // Store C: vgpr r -> row (row0 + r + 8*(lane>=16)), col (n0 + lane%16)
    const int rbase = row0 + ((lane >> 4) << 3);
#pragma unroll
    for (int r = 0; r < 8; ++r) {
        float v = c[r];
        if (RELU) v = fmaxf(v, 0.0f);
        out[(size_t)(rbase + r) * NOUT + n0 + col] = v;
    }
}

// ---------------------------------------------------------------------------
// Host-side orchestration.
// Workspace layout (floats):
//   [0)                : agg   N*128
//   [N*128)            : deg   N
//   [N*128 + N)        : h     N*128   (layer-1 activations)
// ---------------------------------------------------------------------------
extern "C" void kernel_launch(void* const* d_in, const int* in_sizes, int n_in,
                              void* d_out, int out_size, void* d_ws, size_t ws_size,
                              hipStream_t stream) {
    const float*     x    = (const float*)d_in[0];
    const long long* eidx = (const long long*)d_in[1];
    const float*     w1l  = (const float*)d_in[2];
    const float*     b1   = (const float*)d_in[3];
    const float*     w1r  = (const float*)d_in[4];
    const float*     w2l  = (const float*)d_in[5];
    const float*     b2   = (const float*)d_in[6];
    const float*     w2r  = (const float*)d_in[7];
    float*           out  = (float*)d_out;

    float* agg = (float*)d_ws;
    float* deg = agg + (size_t)GN * DIN_;
    float* h   = deg + GN;

    const int zero1_cnt4 = (GN * DIN_ + GN) / 4;  // agg + deg (contiguous)
    const int zero2_cnt4 = (GN * DIN_) / 4;       // agg only
    const int scat_blocks = (GE * 32) / 256;      // wave-per-edge

    // ---- Layer 1 ----
    zero_f32<<<(zero1_cnt4 + 255) / 256, 256, 0, stream>>>((float4*)agg, zero1_cnt4);
    degree_kernel<<<(GE + 255) / 256, 256, 0, stream>>>(eidx, deg, GE);
    scatter_accum<<<scat_blocks, 256, 0, stream>>>(x, eidx, agg, GE);
    sage_gemm<H_, true><<<GN / 16, 256, 0, stream>>>(agg, deg, x, w1l, w1r, b1, h);

    // ---- Layer 2 ----
    zero_f32<<<(zero2_cnt4 + 255) / 256, 256, 0, stream>>>((float4*)agg, zero2_cnt4);
    scatter_accum<<<scat_blocks, 256, 0, stream>>>(h, eidx, agg, GE);
    sage_gemm<DOUT_, false><<<GN / 16, 128, 0, stream>>>(agg, deg, h, w2l, w2r, b2, out);
}